// GNNDecoder_52639119179814
// MI455X (gfx1250) — compile-verified
//
#include <hip/hip_runtime.h>
#include <hip/hip_bf16.h>

typedef float v2f __attribute__((ext_vector_type(2)));
typedef float v4f __attribute__((ext_vector_type(4)));
typedef float v8f __attribute__((ext_vector_type(8)));

#define D 128
#define SA 132   // LDS row stride (floats) for 16x128 activation tiles: 132%64=4 -> conflict-free A frags
#define HS 260   // LDS row stride for 16x256 hid tiles: 260%64=4 -> conflict-free A frags

__device__ __forceinline__ void lds_fence() {
    // order in-wave LDS stores before subsequent LDS fragment loads
    asm volatile("s_wait_dscnt 0" ::: "memory");
}

// ---------------------------------------------------------------------------
// Kernel 0: zero the aggregation buffer (d_out is reused as agg accumulator)
// ---------------------------------------------------------------------------
__global__ __launch_bounds__(256) void k_zero(float* __restrict__ p, int n4) {
    int i = blockIdx.x * 256 + threadIdx.x;
    if (i < n4) {
        *(v4f*)(p + (size_t)i * 4) = (v4f){0.f, 0.f, 0.f, 0.f};
    }
}

// ---------------------------------------------------------------------------
// Kernel 1: h = prelu(x) @ W_enc^T      (one 16-row tile per wave, WMMA f32)
// ---------------------------------------------------------------------------
__global__ __launch_bounds__(128) void k_prelu_gemm(const float* __restrict__ x,
                                                    const float* __restrict__ Wenc,
                                                    const float* __restrict__ prelu_a,
                                                    float* __restrict__ h,
                                                    int N, int numTiles) {
    __shared__ float sA[4][16 * SA];

    const int wave = threadIdx.x >> 5;
    const int lane = threadIdx.x & 31;
    const int lr   = lane & 15;
    const int half = lane >> 4;

    const int tile = blockIdx.x * 4 + wave;
    if (tile >= numTiles) return;                  // wave-uniform: EXEC stays full

    const float slope = prelu_a[0];
    const int   r0    = tile * 16;
    const bool  fullTile = (r0 + 16 <= N);         // wave-uniform
    float* sa = sA[wave];

    // Stage PReLU(x) tile -> LDS (16 rows x 128 cols), float4 per lane-chunk
    for (int i = lane; i < 16 * 32; i += 32) {
        int r = i >> 5;
        int c = (i & 31) << 2;
        int row = r0 + r; if (row >= N) row = N - 1;
        v4f v = *(const v4f*)(x + (size_t)row * D + c);
        v4f p;
        p.x = v.x >= 0.f ? v.x : v.x * slope;
        p.y = v.y >= 0.f ? v.y : v.y * slope;
        p.z = v.z >= 0.f ? v.z : v.z * slope;
        p.w = v.w >= 0.f ? v.w : v.w * slope;
        *(v4f*)(sa + r * SA + c) = p;
    }
    lds_fence();

    // 8 output N-tiles of 16, K = 128 in steps of 4
    for (int nt = 0; nt < 8; ++nt) {
        const int n0 = nt * 16;
        v8f c;
        #pragma unroll
        for (int r = 0; r < 8; ++r) c[r] = 0.f;

        const float* wrow = Wenc + (size_t)(n0 + lr) * D + 2 * half;
        const float* arow = sa + lr * SA + 2 * half;
        #pragma unroll 8
        for (int k = 0; k < D; k += 4) {
            v2f a = *(const v2f*)(arow + k);
            v2f b = *(const v2f*)(wrow + k);
            c = __builtin_amdgcn_wmma_f32_16x16x4_f32(false, a, false, b,
                                                      (short)0, c, false, false);
        }
        float* hrow = h + (size_t)(r0 + 8 * half) * D + n0 + lr;
        if (fullTile) {                            // uniform branch: no exec ladder
            #pragma unroll
            for (int r = 0; r < 8; ++r) hrow[(size_t)r * D] = c[r];
        } else {
            #pragma unroll
            for (int r = 0; r < 8; ++r) {
                int row = r0 + r + 8 * half;
                if (row < N) h[(size_t)row * D + n0 + lr] = c[r];
            }
        }
    }
}

// ---------------------------------------------------------------------------
// Kernel 2: edge scatter: agg[dst] += h[src] + emb1[a0] + emb2[a1]
// one edge per wave; each lane owns a float4 column chunk (coalesced row)
// ---------------------------------------------------------------------------
__global__ __launch_bounds__(256) void k_edge(const float* __restrict__ h,
                                              const int* __restrict__ edge_index,
                                              const int* __restrict__ edge_attr,
                                              const float* __restrict__ emb1,
                                              const float* __restrict__ emb2,
                                              float* agg, int E) {
    const int gid = blockIdx.x * 256 + threadIdx.x;
    const int e   = gid >> 5;
    if (e >= E) return;
    const int c   = (gid & 31) << 2;

    const int src = edge_index[e];
    const int dst = edge_index[E + e];
    const int a0  = edge_attr[2 * e + 0];
    const int a1  = edge_attr[2 * e + 1];

    v4f m  = *(const v4f*)(h    + (size_t)src * D + c);
    v4f e1 = *(const v4f*)(emb1 + (size_t)a0  * D + c);
    v4f e2 = *(const v4f*)(emb2 + (size_t)a1  * D + c);

    float* p = agg + (size_t)dst * D + c;
    atomicAdd(p + 0, m.x + e1.x + e2.x);
    atomicAdd(p + 1, m.y + e1.y + e2.y);
    atomicAdd(p + 2, m.z + e1.z + e2.z);
    atomicAdd(p + 3, m.w + e1.w + e2.w);
}

// ---------------------------------------------------------------------------
// Kernel 3: acc = agg + h + self_emb; hid = relu(acc@W1^T+b1); out = hid@W2^T+b2
// in-place over the agg buffer (per-row independence). One 16-row tile/wave.
// ---------------------------------------------------------------------------
__global__ __launch_bounds__(64) void k_mlp(float* outagg,                 // agg in, out out
                                            const float* __restrict__ h,
                                            const float* __restrict__ emb1,
                                            const float* __restrict__ emb2,
                                            const float* __restrict__ W1,
                                            const float* __restrict__ b1,
                                            const float* __restrict__ W2,
                                            const float* __restrict__ b2,
                                            int N, int numTiles) {
    __shared__ float sAcc[2][16 * SA];
    __shared__ float sHid[2][16 * HS];

    const int wave = threadIdx.x >> 5;
    const int lane = threadIdx.x & 31;
    const int lr   = lane & 15;
    const int half = lane >> 4;

    const int tile = blockIdx.x * 2 + wave;
    if (tile >= numTiles) return;                  // wave-uniform

    const int  r0       = tile * 16;
    const bool fullTile = (r0 + 16 <= N);          // wave-uniform
    float* sa = sAcc[wave];
    float* sh = sHid[wave];

    // Stage acc = agg + h + (emb1[4] + emb2[0]) -> LDS
    for (int i = lane; i < 16 * 32; i += 32) {
        int r = i >> 5;
        int c = (i & 31) << 2;
        int row = r0 + r; if (row >= N) row = N - 1;
        v4f a  = *(const v4f*)(outagg + (size_t)row * D + c);
        v4f hv = *(const v4f*)(h      + (size_t)row * D + c);
        v4f s1 = *(const v4f*)(emb1 + 4 * D + c);
        v4f s2 = *(const v4f*)(emb2 + 0 * D + c);
        v4f acc;
        acc.x = a.x + hv.x + s1.x + s2.x;
        acc.y = a.y + hv.y + s1.y + s2.y;
        acc.z = a.z + hv.z + s1.z + s2.z;
        acc.w = a.w + hv.w + s1.w + s2.w;
        *(v4f*)(sa + r * SA + c) = acc;
    }
    lds_fence();

    // GEMM1: hid(16x256) = acc(16x128) @ W1^T + b1, ReLU; stage hid in LDS
    for (int nt = 0; nt < 16; ++nt) {
        const int n0 = nt * 16;
        const float bias = b1[n0 + lr];
        v8f c;
        #pragma unroll
        for (int r = 0; r < 8; ++r) c[r] = bias;

        const float* wrow = W1 + (size_t)(n0 + lr) * D + 2 * half;
        const float* arow = sa + lr * SA + 2 * half;
        #pragma unroll 8
        for (int k = 0; k < D; k += 4) {
            v2f a = *(const v2f*)(arow + k);
            v2f b = *(const v2f*)(wrow + k);
            c = __builtin_amdgcn_wmma_f32_16x16x4_f32(false, a, false, b,
                                                      (short)0, c, false, false);
        }
        #pragma unroll
        for (int r = 0; r < 8; ++r) {
            float v = c[r] > 0.f ? c[r] : 0.f;     // ReLU
            sh[(r + 8 * half) * HS + n0 + lr] = v;
        }
    }
    lds_fence();

    // GEMM2: out(16x128) = hid(16x256) @ W2^T + b2
    for (int nt = 0; nt < 8; ++nt) {
        const int n0 = nt * 16;
        const float bias = b2[n0 + lr];
        v8f c;
        #pragma unroll
        for (int r = 0; r < 8; ++r) c[r] = bias;

        const float* wrow = W2 + (size_t)(n0 + lr) * 256 + 2 * half;
        const float* arow = sh + lr * HS + 2 * half;
        #pragma unroll 8
        for (int k = 0; k < 256; k += 4) {
            v2f a = *(const v2f*)(arow + k);
            v2f b = *(const v2f*)(wrow + k);
            c = __builtin_amdgcn_wmma_f32_16x16x4_f32(false, a, false, b,
                                                      (short)0, c, false, false);
        }
        float* orow = outagg + (size_t)(r0 + 8 * half) * D + n0 + lr;
        if (fullTile) {                            // uniform branch: no exec ladder
            #pragma unroll
            for (int r = 0; r < 8; ++r) orow[(size_t)r * D] = c[r];
        } else {
            #pragma unroll
            for (int r = 0; r < 8; ++r) {
                int row = r0 + r + 8 * half;
                if (row < N) outagg[(size_t)row * D + n0 + lr] = c[r];
            }
        }
    }
}

// ---------------------------------------------------------------------------
extern "C" void kernel_launch(void* const* d_in, const int* in_sizes, int n_in,
                              void* d_out, int out_size, void* d_ws, size_t ws_size,
                              hipStream_t stream) {
    const float* x          = (const float*)d_in[0];
    const int*   edge_index = (const int*)  d_in[1];
    const int*   edge_attr  = (const int*)  d_in[2];
    const float* prelu_a    = (const float*)d_in[3];
    const float* W_enc      = (const float*)d_in[4];
    const float* emb1       = (const float*)d_in[5];
    const float* emb2       = (const float*)d_in[6];
    const float* W1         = (const float*)d_in[7];
    const float* b1         = (const float*)d_in[8];
    const float* W2         = (const float*)d_in[9];
    const float* b2         = (const float*)d_in[10];

    float* out = (float*)d_out;    // doubles as agg accumulator
    float* h   = (float*)d_ws;     // N*D floats of scratch

    const int N = in_sizes[0] / D;
    const int E = in_sizes[1] / 2;
    const int numTiles = (N + 15) / 16;

    // 1) zero agg (d_out)
    {
        int n4 = (N * D) / 4;
        k_zero<<<(n4 + 255) / 256, 256, 0, stream>>>(out, n4);
    }
    // 2) h = prelu(x) @ W_enc^T
    {
        int blocks = (numTiles + 3) / 4;           // 4 waves/block, 1 tile/wave
        k_prelu_gemm<<<blocks, 128, 0, stream>>>(x, W_enc, prelu_a, h, N, numTiles);
    }
    // 3) edge scatter-add into agg
    {
        long long threads = (long long)E * 32;
        int blocks = (int)((threads + 255) / 256);
        k_edge<<<blocks, 256, 0, stream>>>(h, edge_index, edge_attr, emb1, emb2, out, E);
    }
    // 4) fused self-loop + GIN MLP, in place over d_out
    {
        int blocks = (numTiles + 1) / 2;           // 2 waves/block, 1 tile/wave
        k_mlp<<<blocks, 64, 0, stream>>>(out, h, emb1, emb2, W1, b1, W2, b2, N, numTiles);
    }
}